// RandlaNet_mlp_17154099380545
// MI455X (gfx1250) — compile-verified
//
#include <hip/hip_runtime.h>
#include <hip/hip_bf16.h>

// ---------------------------------------------------------------------------
// RandLA-Net forward for MI455X (gfx1250).  All linear algebra runs through
// V_WMMA_F32_16X16X4_F32 (f32 A/B, f32 accum, K=4) so the matrix pipes carry
// both the SharedMLP GEMMs and the KNN distance cross-terms.
// ---------------------------------------------------------------------------

typedef float  v2f __attribute__((ext_vector_type(2)));
typedef float  v8f __attribute__((ext_vector_type(8)));

#define LRELU 0.2f
#define BN_EPS 1e-5f
#define KNN_K 16

__device__ __forceinline__ v8f wmma_f32x4(v2f a, v2f b, v8f c) {
  return __builtin_amdgcn_wmma_f32_16x16x4_f32(false, a, false, b, (short)0, c,
                                               false, false);
}

// ---------------------------------------------------------------------------
// Fused GEMM + epilogue:  Y[M,N] = act(bn(X[M,K] @ W[K,N] + b))
// flags: 1 = +bias, 2 = batchnorm(eval), 4 = LeakyReLU(0.2)
// One wave -> one 32x16 output tile (two stacked 16x16 WMMA tiles sharing the
// B fragment).  Out-of-range rows/cols are CLAMPED, not masked: clamped lanes
// only feed accumulator entries that are never stored, so the K-loop is fully
// branch-free.  UNSIGNED 32-bit indices off uniform bases -> provable zext ->
// saddr + voffset (scale_offset) addressing, no 64-bit VALU address math.
// Explicit 1-deep software pipeline: fragments for step k+1 are issued before
// the WMMAs of step k, so s_wait_loadcnt lands a full iteration behind.
// A layout: lane l<16 holds A[M=l][K=0..1], lane 16+l holds A[M=l][K=2..3].
// B layout: lane l<16 holds B[K=0..1][N=l], lane 16+l holds B[K=2..3][N=l-16].
// C/D: VGPR r, lanes 0-15 -> D[M=r][N=lane], lanes 16-31 -> D[M=8+r][N=l-16].
// ---------------------------------------------------------------------------
__global__ __launch_bounds__(128) void gemm_ep_kernel(
    const float* __restrict__ X, const float* __restrict__ W,
    const float* __restrict__ bias,
    const float* __restrict__ gamma, const float* __restrict__ beta,
    const float* __restrict__ mean,  const float* __restrict__ var,
    float* __restrict__ Y, int M, int K, int N, int flags) {
  const int lane = threadIdx.x & 31;
  const int wv   = threadIdx.x >> 5;
  const int tiles_n = (N + 15) >> 4;
  const int tiles_m = (M + 31) >> 5;              // 32 rows per wave
  const int tile = blockIdx.x * 4 + wv;
  if (tile >= tiles_m * tiles_n) return;          // whole-wave exit: EXEC stays full
  const int tm = tile / tiles_n;
  const int tn = tile % tiles_n;

  const int row0 = tm * 32 + (lane & 15);
  const int col  = tn * 16 + (lane & 15);
  const int koff = (lane >> 4) << 1;              // 0 or 2

  // clamped indices: safe because clamped lanes land in never-stored D slots
  const int r0c = min(row0, M - 1);
  const int r1c = min(row0 + 16, M - 1);
  const int cc  = min(col, N - 1);

  // unsigned 32-bit element offsets (all tensors < 2^31 elements) -> zext addr
  const unsigned uN = (unsigned)N;
  unsigned xi0 = (unsigned)(r0c * K + koff);
  unsigned xi1 = (unsigned)(r1c * K + koff);
  unsigned wi  = (unsigned)(koff * N + cc);

  v8f acc0 = {};
  v8f acc1 = {};
  const int K4 = K & ~3;
  if (K4 > 0) {
    v2f a0, a1, b;
    a0.x = X[xi0]; a0.y = X[xi0 + 1u];
    a1.x = X[xi1]; a1.y = X[xi1 + 1u];
    b.x  = W[wi];  b.y  = W[wi + uN];
    for (int k = 4; k < K4; k += 4) {
      xi0 += 4u; xi1 += 4u; wi += 4u * uN;
      v2f na0, na1, nb;                 // prefetch next fragments first
      na0.x = X[xi0]; na0.y = X[xi0 + 1u];
      na1.x = X[xi1]; na1.y = X[xi1 + 1u];
      nb.x  = W[wi];  nb.y  = W[wi + uN];
      acc0 = wmma_f32x4(a0, b, acc0);
      acc1 = wmma_f32x4(a1, b, acc1);
      a0 = na0; a1 = na1; b = nb;
    }
    acc0 = wmma_f32x4(a0, b, acc0);
    acc1 = wmma_f32x4(a1, b, acc1);
    xi0 += 4u; xi1 += 4u; wi += 4u * uN;
  }
  if (K4 < K) {                                   // tail (K % 4 != 0: K = 6, 10)
    const int ka = K4 + koff;
    // clamp deltas (<= 0); unsigned wrap is defined and results stay in range
    const unsigned d0 = (unsigned)(min(ka, K - 1) - ka);
    const unsigned d1 = (unsigned)(min(ka + 1, K - 1) - ka);
    v2f a0, a1, b;
    a0.x = X[xi0 + d0];  a0.y = X[xi0 + d1];
    a1.x = X[xi1 + d0];  a1.y = X[xi1 + d1];
    b.x = (ka     < K) ? W[wi + d0 * uN] : 0.f;   // mask product via B only
    b.y = (ka + 1 < K) ? W[wi + d1 * uN] : 0.f;
    acc0 = wmma_f32x4(a0, b, acc0);
    acc1 = wmma_f32x4(a1, b, acc1);
  }

  if (col < N) {
    float bb = 0.f, g = 1.f, bt = 0.f, mn = 0.f, iv = 1.f;
    if (flags & 1) bb = bias[col];
    if (flags & 2) {
      g = gamma[col]; bt = beta[col]; mn = mean[col];
      iv = rsqrtf(var[col] + BN_EPS);
    }
    const int mbase = tm * 32 + ((lane >> 4) << 3);
#pragma unroll
    for (int r = 0; r < 8; ++r) {
      const int rr0 = mbase + r;
      if (rr0 < M) {
        float v = acc0[r] + bb;
        if (flags & 2) v = (v - mn) * iv * g + bt;
        if (flags & 4) v = (v >= 0.f) ? v : LRELU * v;
        Y[(unsigned)(rr0 * N + col)] = v;
      }
      const int rr1 = rr0 + 16;
      if (rr1 < M) {
        float v = acc1[r] + bb;
        if (flags & 2) v = (v - mn) * iv * g + bt;
        if (flags & 4) v = (v >= 0.f) ? v : LRELU * v;
        Y[(unsigned)(rr1 * N + col)] = v;
      }
    }
  }
}

// ---------------------------------------------------------------------------
// KNN (k=16, loop=true, queries==refs).  One wave handles 16 query rows.
// Distance proxy d = |r|^2 - 2 q.r  (query-constant |q|^2 dropped) computed
// per 16x16 tile with a single WMMA f32 16x16x4:  A=(-2qx,-2qy,-2qz,1),
// B rows=(rx,ry,rz,|r|^2).  Tile staged through LDS; lanes 0..15 each keep a
// sorted top-16 (insertion + single bubble pass) in registers.
// ---------------------------------------------------------------------------
__global__ __launch_bounds__(32) void knn16_kernel(
    const float* __restrict__ pos, int n, int* __restrict__ nbr) {
  __shared__ float dtile[16][17];
  const int lane = threadIdx.x;
  const int qrow = blockIdx.x * 16 + (lane & 15);
  const int koff = (lane >> 4) << 1;
  const bool qok = qrow < n;

  const int qc = min(qrow, n - 1);
  const float qx = pos[qc * 3], qy = pos[qc * 3 + 1], qz = pos[qc * 3 + 2];
  v2f a;
  a.x = (koff == 0) ? -2.f * qx : -2.f * qz;
  a.y = (koff == 0) ? -2.f * qy : 1.0f;

  float bestd[KNN_K];
  int   besti[KNN_K];
#pragma unroll
  for (int s = 0; s < KNN_K; ++s) { bestd[s] = 3.4e38f; besti[s] = 0; }

  const int ntiles = (n + 15) >> 4;
  for (int rt = 0; rt < ntiles; ++rt) {
    const int rcol = rt * 16 + (lane & 15);
    const int rc = min(rcol, n - 1);
    const float rx = pos[rc * 3], ry = pos[rc * 3 + 1], rz = pos[rc * 3 + 2];
    const float rn2 = rx * rx + ry * ry + rz * rz;
    v2f b;
    b.x = (koff == 0) ? rx : rz;
    b.y = (koff == 0) ? ry : rn2;

    v8f acc = {};
    acc = wmma_f32x4(a, b, acc);
    const int m0 = (lane >> 4) << 3;
    const int nc = lane & 15;
#pragma unroll
    for (int r = 0; r < 8; ++r) dtile[m0 + r][nc] = acc[r];
    __syncthreads();

    if (lane < 16 && qok) {
#pragma unroll
      for (int j = 0; j < 16; ++j) {
        const int ridx = rt * 16 + j;
        const float d = dtile[lane][j];
        if (ridx < n && d < bestd[KNN_K - 1]) {
          bestd[KNN_K - 1] = d; besti[KNN_K - 1] = ridx;
#pragma unroll
          for (int s = KNN_K - 1; s > 0; --s) {
            if (bestd[s] < bestd[s - 1]) {
              float td = bestd[s]; bestd[s] = bestd[s - 1]; bestd[s - 1] = td;
              int   ti = besti[s]; besti[s] = besti[s - 1]; besti[s - 1] = ti;
            }
          }
        }
      }
    }
    __syncthreads();
  }
  if (lane < 16 && qok) {
#pragma unroll
    for (int s = 0; s < KNN_K; ++s) nbr[qrow * KNN_K + s] = besti[s];
  }
}

// ---------------------------------------------------------------------------
// Relative point-position encoding: [pos_i, pos_j, rel, |rel|]  -> [n*16, 10]
// ---------------------------------------------------------------------------
__global__ __launch_bounds__(256) void relinfo_kernel(
    const float* __restrict__ pos, const int* __restrict__ nbr, int n,
    float* __restrict__ out) {
  const int e = blockIdx.x * blockDim.x + threadIdx.x;
  if (e >= n * KNN_K) return;
  const int i = e >> 4;
  const int j = nbr[e];
  const float xi = pos[i * 3], yi = pos[i * 3 + 1], zi = pos[i * 3 + 2];
  const float xj = pos[j * 3], yj = pos[j * 3 + 1], zj = pos[j * 3 + 2];
  const float rx = xi - xj, ry = yi - yj, rz = zi - zj;
  const float d = sqrtf(rx * rx + ry * ry + rz * rz);
  float* o = out + (long long)e * 10;
  o[0] = xi; o[1] = yi; o[2] = zi; o[3] = xj; o[4] = yj; o[5] = zj;
  o[6] = rx; o[7] = ry; o[8] = rz; o[9] = d;
}

// local[e] = [ x[nbr[e]] (c1) , enc[e] (c2) ]
__global__ __launch_bounds__(256) void gather_concat_edges_kernel(
    const float* __restrict__ x, int c1, const float* __restrict__ enc, int c2,
    const int* __restrict__ nbr, int n, float* __restrict__ out) {
  const int c = c1 + c2;
  const long long i = (long long)blockIdx.x * blockDim.x + threadIdx.x;
  const long long total = (long long)n * KNN_K * c;
  if (i >= total) return;
  const int ch = (int)(i % c);
  const long long e = i / c;
  float v;
  if (ch < c1) v = x[(long long)nbr[e] * c1 + ch];
  else         v = enc[e * c2 + (ch - c1)];
  out[i] = v;
}

// Attentive pooling: out[n,c] = sum_k softmax_k(att[n,k,c]) * local[n,k,c]
__global__ __launch_bounds__(256) void attn_pool_kernel(
    const float* __restrict__ att, const float* __restrict__ local,
    int n, int c, float* __restrict__ out) {
  const int i = blockIdx.x * blockDim.x + threadIdx.x;
  if (i >= n * c) return;
  const int pt = i / c, ch = i % c;
  const float* ap = att   + (long long)(pt * KNN_K) * c + ch;
  const float* lp = local + (long long)(pt * KNN_K) * c + ch;
  float av[KNN_K];
  float m = -3.4e38f;
#pragma unroll
  for (int k = 0; k < KNN_K; ++k) { av[k] = ap[(long long)k * c]; m = fmaxf(m, av[k]); }
  float Z = 0.f, s = 0.f;
#pragma unroll
  for (int k = 0; k < KNN_K; ++k) {
    const float e = __expf(av[k] - m);
    Z += e;
    s += e * lp[(long long)k * c];
  }
  out[i] = s / Z;
}

__global__ __launch_bounds__(256) void add_leaky_kernel(
    const float* __restrict__ a, const float* __restrict__ b, long long n,
    float* __restrict__ out) {
  const long long i = (long long)blockIdx.x * blockDim.x + threadIdx.x;
  if (i >= n) return;
  const float v = a[i] + b[i];
  out[i] = (v >= 0.f) ? v : LRELU * v;
}

// out[i,:] = in[idx[i],:]
__global__ __launch_bounds__(256) void gather_rows_kernel(
    const float* __restrict__ in, const int* __restrict__ idx, int m, int c,
    float* __restrict__ out) {
  const long long i = (long long)blockIdx.x * blockDim.x + threadIdx.x;
  if (i >= (long long)m * c) return;
  const int r = (int)(i / c), ch = (int)(i % c);
  out[i] = in[(long long)idx[r] * c + ch];
}

// knn_interpolate(k=1): per query argmin over refs, then concat features.
__global__ __launch_bounds__(128) void nn_gather_concat_kernel(
    const float* __restrict__ qpos, int nq, const float* __restrict__ rpos, int nr,
    const float* __restrict__ xr, int c1, const float* __restrict__ xskip, int c2,
    float* __restrict__ out) {
  const int q = blockIdx.x * blockDim.x + threadIdx.x;
  if (q >= nq) return;
  const float qx = qpos[q * 3], qy = qpos[q * 3 + 1], qz = qpos[q * 3 + 2];
  int best = 0; float bd = 3.4e38f;
  for (int r = 0; r < nr; ++r) {
    const float dx = qx - rpos[r * 3];
    const float dy = qy - rpos[r * 3 + 1];
    const float dz = qz - rpos[r * 3 + 2];
    const float d = dx * dx + dy * dy + dz * dz;
    if (d < bd) { bd = d; best = r; }
  }
  float* o = out + (long long)q * (c1 + c2);
  for (int ch = 0; ch < c1; ++ch) o[ch] = xr[(long long)best * c1 + ch];
  for (int ch = 0; ch < c2; ++ch) o[c1 + ch] = xskip[(long long)q * c2 + ch];
}

__global__ __launch_bounds__(256) void log_softmax_kernel(
    const float* __restrict__ in, int n, int c, float* __restrict__ out) {
  const int i = blockIdx.x * blockDim.x + threadIdx.x;
  if (i >= n) return;
  const float* r = in + (long long)i * c;
  float m = -3.4e38f;
  for (int j = 0; j < c; ++j) m = fmaxf(m, r[j]);
  float Z = 0.f;
  for (int j = 0; j < c; ++j) Z += __expf(r[j] - m);
  const float lz = m + __logf(Z);
  float* o = out + (long long)i * c;
  for (int j = 0; j < c; ++j) o[j] = r[j] - lz;
}

__global__ __launch_bounds__(128) void seed_kernel(
    const int* __restrict__ idx1, const int* __restrict__ idx2,
    const int* __restrict__ idx3, const int* __restrict__ idx4, int n4,
    float* __restrict__ out) {
  const int i = blockIdx.x * blockDim.x + threadIdx.x;
  if (i >= n4) return;
  int s = idx4[i];
  s = idx3[s];
  s = idx2[s];
  s = idx1[s];
  out[i] = (float)s;
}

// ---------------------------------------------------------------------------
// Host orchestration
// ---------------------------------------------------------------------------
namespace {

inline int cdiv(long long a, long long b) { return (int)((a + b - 1) / b); }

struct Arena {
  char* base; size_t off; size_t cap;
  void* alloc(size_t bytes) {
    size_t a = (off + 255) & ~(size_t)255;
    off = a + bytes;
    return base + a;
  }
  float* af(size_t n) { return (float*)alloc(n * sizeof(float)); }
  int*   ai(size_t n) { return (int*)alloc(n * sizeof(int)); }
  size_t mark() const { return off; }
  void reset(size_t m) { off = m; }
};

struct Layer {  // leaf order (jax tree, sorted keys): W, b, beta, gamma, mean, var
  const float *W, *b, *beta, *gamma, *mean, *var;
  int din, dout;
};
struct Lfa {    // att_W, encoder(list[1]), post(list[1])
  const float* attW;
  Layer enc, post;
  int c;
};
struct Block {  // lfa1, lfa2, mlp1, mlp2, shortcut
  Lfa lfa1, lfa2;
  Layer mlp1, mlp2, shortcut;
  int din, dout;
};

void launch_gemm(hipStream_t st, const float* X, int M, int K, const float* W, int N,
                 const float* bias, const float* gamma, const float* beta,
                 const float* mean, const float* var, int flags, float* Y) {
  const int tiles = cdiv(M, 32) * cdiv(N, 16);
  gemm_ep_kernel<<<cdiv(tiles, 4), 128, 0, st>>>(X, W, bias, gamma, beta, mean, var,
                                                 Y, M, K, N, flags);
}

void run_layer(hipStream_t st, const Layer& L, const float* X, int M, float* Y, bool act) {
  launch_gemm(st, X, M, L.din, L.W, L.dout, L.b, L.gamma, L.beta, L.mean, L.var,
              1 | 2 | (act ? 4 : 0), Y);
}

void run_lfa(hipStream_t st, Arena& A, const Lfa& L, const float* h /*[n,c/2]*/,
             const float* pos, const int* nbr, int n, float* out /*[n,c]*/) {
  const int c = L.c, ch = c / 2;
  const long long E = (long long)n * KNN_K;
  const size_t mk = A.mark();

  float* enc = A.af(E * ch);
  {
    const size_t m1 = A.mark();
    float* rel = A.af(E * 10);
    relinfo_kernel<<<cdiv(E, 256), 256, 0, st>>>(pos, nbr, n, rel);
    run_layer(st, L.enc, rel, (int)E, enc, true);   // 10 -> c/2, bn + leaky
    A.reset(m1);
  }
  float* local = A.af(E * c);
  gather_concat_edges_kernel<<<cdiv(E * c, 256), 256, 0, st>>>(h, ch, enc, ch, nbr, n,
                                                               local);
  float* att = A.af(E * c);
  launch_gemm(st, local, (int)E, c, L.attW, c, nullptr, nullptr, nullptr, nullptr,
              nullptr, 0, att);                     // bias=False, no norm/act
  float* pooled = A.af((size_t)n * c);
  attn_pool_kernel<<<cdiv((long long)n * c, 256), 256, 0, st>>>(att, local, n, c,
                                                                pooled);
  run_layer(st, L.post, pooled, n, out, true);      // c -> c
  A.reset(mk);
}

void run_block(hipStream_t st, Arena& A, const Block& B, const float* x,
               const float* pos, int n, float* out /*[n,dout]*/) {
  const size_t mk = A.mark();
  int* nbr = A.ai((size_t)n * KNN_K);
  knn16_kernel<<<cdiv(n, 16), 32, 0, st>>>(pos, n, nbr);

  float* sc = A.af((size_t)n * B.dout);
  run_layer(st, B.shortcut, x, n, sc, false);

  float* hA = A.af((size_t)n * B.dout);
  float* hB = A.af((size_t)n * B.dout);
  run_layer(st, B.mlp1, x, n, hA, true);            // [n, dout/8]
  run_lfa(st, A, B.lfa1, hA, pos, nbr, n, hB);      // [n, dout/4]
  run_lfa(st, A, B.lfa2, hB, pos, nbr, n, hA);      // [n, dout/2]
  run_layer(st, B.mlp2, hA, n, hB, false);          // [n, dout]
  add_leaky_kernel<<<cdiv((long long)n * B.dout, 256), 256, 0, st>>>(
      hB, sc, (long long)n * B.dout, out);
  A.reset(mk);
}

void run_fp(hipStream_t st, Arena& A, const Layer& L, const float* x,
            const float* pos_r, int nr, const float* xskip, const float* pos_q,
            int nq, int c1, int c2, float* y /*[nq, L.dout]*/) {
  const size_t mk = A.mark();
  float* cat = A.af((size_t)nq * (c1 + c2));
  nn_gather_concat_kernel<<<cdiv(nq, 128), 128, 0, st>>>(pos_q, nq, pos_r, nr, x, c1,
                                                         xskip, c2, cat);
  run_layer(st, L, cat, nq, y, true);
  A.reset(mk);
}

}  // namespace

extern "C" void kernel_launch(void* const* d_in, const int* in_sizes, int n_in,
                              void* d_out, int out_size, void* d_ws, size_t ws_size,
                              hipStream_t stream) {
  const float* x    = (const float*)d_in[0];
  const float* pos  = (const float*)d_in[1];
  // d_in[2] = batch (all-zero, unused)
  const int* idx1 = (const int*)d_in[3];
  const int* idx2 = (const int*)d_in[4];
  const int* idx3 = (const int*)d_in[5];
  const int* idx4 = (const int*)d_in[6];

  const int N0 = in_sizes[0] / 6;
  const int n1 = in_sizes[3];
  const int n2 = in_sizes[4];
  const int n3 = in_sizes[5];
  const int n4 = in_sizes[6];

  // ---- parse params (jax tree leaf order: dict keys sorted, lists in order) ----
  int p = 7;
  auto nextp = [&]() { return (const float*)d_in[p++]; };
  auto layer = [&](int din, int dout) {
    Layer L;
    L.W = nextp(); L.b = nextp(); L.beta = nextp(); L.gamma = nextp();
    L.mean = nextp(); L.var = nextp();
    L.din = din; L.dout = dout;
    return L;
  };
  auto lfa = [&](int c) {
    Lfa f;
    f.attW = nextp();
    f.enc  = layer(10, c / 2);
    f.post = layer(c, c);
    f.c = c;
    return f;
  };
  auto block = [&](int din, int dout) {
    Block b;
    b.lfa1 = lfa(dout / 4);
    b.lfa2 = lfa(dout / 2);
    b.mlp1 = layer(din, dout / 8);
    b.mlp2 = layer(dout / 2, dout);
    b.shortcut = layer(din, dout);
    b.din = din; b.dout = dout;
    return b;
  };

  // sorted top-level: block1..4, fc0, fc_classif, fc_inst, inst_fp1..4,
  // mlp_classif, mlp_summit, sem_fp1..4
  Block B1 = block(8, 32);
  Block B2 = block(32, 128);
  Block B3 = block(128, 256);
  Block B4 = block(256, 512);
  const float* fc0W = nextp();     const float* fc0b = nextp();
  const float* fccW = nextp();     const float* fccb = nextp();
  const float* fciW = nextp();     const float* fcib = nextp();
  Layer ifp1 = layer(64, 8);
  Layer ifp2 = layer(160, 32);
  Layer ifp3 = layer(384, 128);
  Layer ifp4 = layer(768, 256);
  Layer cls0 = layer(8, 64);
  Layer cls1 = layer(64, 32);
  Layer summ = layer(512, 512);
  Layer sfp1 = layer(64, 8);
  Layer sfp2 = layer(160, 32);
  Layer sfp3 = layer(384, 128);
  Layer sfp4 = layer(768, 256);

  Arena A{(char*)d_ws, 0, ws_size};
  hipStream_t st = stream;

  // ---- encoder ----
  float* h0 = A.af((size_t)N0 * 8);
  launch_gemm(st, x, N0, 6, fc0W, 8, fc0b, nullptr, nullptr, nullptr, nullptr, 1, h0);

  float* b1 = A.af((size_t)N0 * 32);
  run_block(st, A, B1, h0, pos, N0, b1);
  float* pos1 = A.af((size_t)n1 * 3);
  gather_rows_kernel<<<cdiv((long long)n1 * 3, 256), 256, 0, st>>>(pos, idx1, n1, 3, pos1);
  float* b1d = A.af((size_t)n1 * 32);
  gather_rows_kernel<<<cdiv((long long)n1 * 32, 256), 256, 0, st>>>(b1, idx1, n1, 32, b1d);

  float* b2 = A.af((size_t)n1 * 128);
  run_block(st, A, B2, b1d, pos1, n1, b2);
  float* pos2 = A.af((size_t)n2 * 3);
  gather_rows_kernel<<<cdiv((long long)n2 * 3, 256), 256, 0, st>>>(pos1, idx2, n2, 3, pos2);
  float* b2d = A.af((size_t)n2 * 128);
  gather_rows_kernel<<<cdiv((long long)n2 * 128, 256), 256, 0, st>>>(b2, idx2, n2, 128, b2d);

  float* b3 = A.af((size_t)n2 * 256);
  run_block(st, A, B3, b2d, pos2, n2, b3);
  float* pos3 = A.af((size_t)n3 * 3);
  gather_rows_kernel<<<cdiv((long long)n3 * 3, 256), 256, 0, st>>>(pos2, idx3, n3, 3, pos3);
  float* b3d = A.af((size_t)n3 * 256);
  gather_rows_kernel<<<cdiv((long long)n3 * 256, 256), 256, 0, st>>>(b3, idx3, n3, 256, b3d);

  float* b4 = A.af((size_t)n3 * 512);
  run_block(st, A, B4, b3d, pos3, n3, b4);
  float* pos4 = A.af((size_t)n4 * 3);
  gather_rows_kernel<<<cdiv((long long)n4 * 3, 256), 256, 0, st>>>(pos3, idx4, n4, 3, pos4);
  float* b4d = A.af((size_t)n4 * 512);
  gather_rows_kernel<<<cdiv((long long)n4 * 512, 256), 256, 0, st>>>(b4, idx4, n4, 512, b4d);

  float* summit = A.af((size_t)n4 * 512);
  run_layer(st, summ, b4d, n4, summit, true);

  float* sem_out  = (float*)d_out;                       // [N0, 20]
  float* inst_out = (float*)d_out + (size_t)N0 * 20;     // [N0, 32]
  float* seed_out = (float*)d_out + (size_t)N0 * 52;     // [n4]

  // ---- semantic decoder ----
  {
    const size_t mk = A.mark();
    float* y4 = A.af((size_t)n3 * 256);
    run_fp(st, A, sfp4, summit, pos4, n4, b3d, pos3, n3, 512, 256, y4);
    float* y3 = A.af((size_t)n2 * 128);
    run_fp(st, A, sfp3, y4, pos3, n3, b2d, pos2, n2, 256, 128, y3);
    float* y2 = A.af((size_t)n1 * 32);
    run_fp(st, A, sfp2, y3, pos2, n2, b1d, pos1, n1, 128, 32, y2);
    float* y1 = A.af((size_t)N0 * 8);
    run_fp(st, A, sfp1, y2, pos1, n1, b1, pos, N0, 32, 32, y1);
    float* hid = A.af((size_t)N0 * 64);
    run_layer(st, cls0, y1, N0, hid, true);
    float* hid2 = A.af((size_t)N0 * 32);
    run_layer(st, cls1, hid, N0, hid2, true);
    float* logits = A.af((size_t)N0 * 20);
    launch_gemm(st, hid2, N0, 32, fccW, 20, fccb, nullptr, nullptr, nullptr, nullptr,
                1, logits);
    log_softmax_kernel<<<cdiv(N0, 256), 256, 0, st>>>(logits, N0, 20, sem_out);
    A.reset(mk);
  }

  // ---- instance decoder ----
  {
    const size_t mk = A.mark();
    float* y4 = A.af((size_t)n3 * 256);
    run_fp(st, A, ifp4, summit, pos4, n4, b3d, pos3, n3, 512, 256, y4);
    float* y3 = A.af((size_t)n2 * 128);
    run_fp(st, A, ifp3, y4, pos3, n3, b2d, pos2, n2, 256, 128, y3);
    float* y2 = A.af((size_t)n1 * 32);
    run_fp(st, A, ifp2, y3, pos2, n2, b1d, pos1, n1, 128, 32, y2);
    float* y1 = A.af((size_t)N0 * 8);
    run_fp(st, A, ifp1, y2, pos1, n1, b1, pos, N0, 32, 32, y1);
    launch_gemm(st, y1, N0, 8, fciW, 32, fcib, nullptr, nullptr, nullptr, nullptr, 1,
                inst_out);
    A.reset(mk);
  }

  // ---- seed indices ----
  seed_kernel<<<cdiv(n4, 128), 128, 0, st>>>(idx1, idx2, idx3, idx4, n4, seed_out);
}